// S3Pure_6519760355898
// MI455X (gfx1250) — compile-verified
//
#include <hip/hip_runtime.h>
#include <math.h>

// ---------------------------------------------------------------------------
// S3Pure quaternion scan for MI455X (gfx1250).
//
// Roofline: ~0.1 GFLOP, ~35 MB moved -> memory/latency bound (HBM floor
// ~1.5us @ 23.3 TB/s; embed table 8 MB fits in 192 MB L2). WMMA does not
// apply (no dense matmul); the CDNA5 features that matter are the Tensor
// Data Mover *gather mode* for the token->embedding lookup (async, tracked
// by TENSORcnt) and wave32 lane reductions.
//
// Algorithm (quaternion product is associative; per-step l2-normalize of a
// product of unit quaternions is numerical hygiene only, so chunked scan is
// valid):
//   phase1: chunk products  P[b,m,k] = prod_{t in chunk k} g_t   (131072 thr)
//   phase2: exclusive scan  pre[b,m,k], and C_final               (512 thr)
//   phase3: replay chunks from pre[], emit sigma with 16-lane reduction
// ---------------------------------------------------------------------------

#define B   32
#define T   4096
#define V   32000
#define M   16
#define CL  32              // steps per chunk
#define NC  (T / CL)        // 128 chunks per chain
#define CPB 8               // chunks per block
#define STEPS_PB (CPB * CL) // 256 timesteps staged per block (64 KB LDS tile)

struct Q { float w, x, y, z; };

__device__ __forceinline__ Q qmul(Q a, Q b) {
    Q r;
    r.w = a.w * b.w - a.x * b.x - a.y * b.y - a.z * b.z;
    r.x = a.w * b.x + a.x * b.w + a.y * b.z - a.z * b.y;
    r.y = a.w * b.y - a.x * b.z + a.y * b.w + a.z * b.x;
    r.z = a.w * b.z + a.x * b.y - a.y * b.x + a.z * b.w;
    return r;
}

__device__ __forceinline__ Q qnormalize(Q a) {
    float n = sqrtf(a.w * a.w + a.x * a.x + a.y * a.y + a.z * a.z);
    float inv = 1.0f / fmaxf(n, 1e-12f);
    Q r; r.w = a.w * inv; r.x = a.x * inv; r.y = a.y * inv; r.z = a.z * inv;
    return r;
}

// ------------------------- TDM gather staging ------------------------------
#if defined(__has_builtin)
#if __has_builtin(__builtin_amdgcn_tensor_load_to_lds)
#define HAVE_TDM 1
#endif
#endif
#ifndef HAVE_TDM
#define HAVE_TDM 0
#endif

#if HAVE_TDM
typedef unsigned int u32x4 __attribute__((ext_vector_type(4)));
typedef int          i32x4 __attribute__((ext_vector_type(4)));
typedef int          i32x8 __attribute__((ext_vector_type(8)));

// Issue one gather-mode TDM op: 16 embedding rows (16-bit row indices,
// 64 f32 = 256 B each) -> LDS at lds_byte_off. D# layout per CDNA5 ISA §8.
__device__ __forceinline__ void tdm_gather16(const float* embed,
                                             const int* tok_ptr,
                                             unsigned lds_byte_off) {
    int d[8];
#pragma unroll
    for (int p = 0; p < 8; ++p) {
        int lo = tok_ptr[2 * p] & 0xFFFF;
        int hi = tok_ptr[2 * p + 1] & 0xFFFF;
        d[p] = __builtin_amdgcn_readfirstlane(lo | (hi << 16));
    }
    unsigned long long ga = (unsigned long long)(const char*)embed;
    unsigned ga_lo = (unsigned)__builtin_amdgcn_readfirstlane((int)(ga & 0xFFFFFFFFu));
    unsigned ga_hi = (unsigned)__builtin_amdgcn_readfirstlane((int)(ga >> 32));

    // group0: count=1 | gather_mode(b31) | 16-bit idx(b30=0); lds_addr; 57b
    // global addr; type=2 in bits[127:126].
    u32x4 g0 = { 0x80000001u,
                 lds_byte_off,
                 ga_lo,
                 (ga_hi & 0x01FFFFFFu) | 0x80000000u };
    // group1: data_size=4B(code 2)<<16 ; tensor_dim0=64 ; tensor_dim1=V ;
    // tile_dim0=64 ; tile_dim1=16 valid indices ; tensor_dim0_stride=64.
    i32x8 g1 = { (int)(2u << 16),
                 (int)(64u << 16),
                 (int)((unsigned)V << 16),
                 (int)(64u << 16),
                 16,
                 64,
                 0, 0 };
    i32x4 g2 = { d[0], d[1], d[2], d[3] };   // row indices 0..7
    i32x4 g3 = { d[4], d[5], d[6], d[7] };   // row indices 8..15
#if __clang_major__ >= 23
    i32x8 zero = { 0, 0, 0, 0, 0, 0, 0, 0 };
    __builtin_amdgcn_tensor_load_to_lds(g0, g1, g2, g3, zero, 0);
#else
    __builtin_amdgcn_tensor_load_to_lds(g0, g1, g2, g3, 0);
#endif
}
#endif // HAVE_TDM

// Stage STEPS_PB gathered embedding rows (64 f32 each) into lds (offset 0 of
// the kernel's single __shared__ block).
__device__ __forceinline__ void stage_block(const int* tokens_bt0,
                                            const float* embed,
                                            float* lds) {
#if HAVE_TDM
    if (threadIdx.x < 32) { // wave 0 issues all TDM ops
#pragma unroll 1
        for (int j = 0; j < STEPS_PB / 16; ++j)
            tdm_gather16(embed, tokens_bt0 + j * 16, (unsigned)(j * 16 * 256));
        __builtin_amdgcn_s_wait_tensorcnt(0);
    }
    // The TDM writes LDS behind the compiler's back (the builtin carries no
    // pointer). Round-1 histogram showed ds=0: with no visible store to
    // lds_g, its loads folded to undef and the compute chain was dead-coded.
    // Escape the pointer through an opaque asm with a memory clobber so the
    // ds_load_b128s below survive.
    asm volatile("" : "+v"(lds) :: "memory");
    __syncthreads();
#else
    for (int i = (int)threadIdx.x; i < STEPS_PB * 64; i += (int)blockDim.x) {
        int tok = tokens_bt0[i >> 6];
        lds[i] = embed[tok * 64 + (i & 63)];
    }
    __syncthreads();
#endif
}

// --------------------------------- phase 1 ---------------------------------
__global__ __launch_bounds__(256) void s3_phase1(const int* __restrict__ tokens,
                                                 const float* __restrict__ embed,
                                                 float* __restrict__ wsP) {
    __shared__ float lds_g[STEPS_PB * 64]; // 64 KB
    const int b  = (int)blockIdx.x & (B - 1);
    const int cg = (int)blockIdx.x >> 5;
    const int t0 = cg * STEPS_PB;
    const int m  = (int)threadIdx.x & (M - 1);
    const int c  = (int)threadIdx.x >> 4;

    stage_block(tokens + b * T + t0, embed, lds_g);

    Q P = {1.f, 0.f, 0.f, 0.f};
    const float* row = lds_g + (c * CL) * 64 + m * 4;
#pragma unroll 4
    for (int i = 0; i < CL; ++i) {
        float4 gv = *(const float4*)(row + i * 64);   // ds_load_b128
        Q g; g.w = gv.x; g.x = gv.y; g.y = gv.z; g.z = gv.w;
        P = qnormalize(qmul(P, qnormalize(g)));
    }
    const int chunk = cg * CPB + c;
    float4 o; o.x = P.w; o.y = P.x; o.z = P.y; o.w = P.z;
    *(float4*)(wsP + (((b * M + m) * NC) + chunk) * 4) = o;
}

// --------------------------------- phase 2 ---------------------------------
__global__ __launch_bounds__(256) void s3_phase2(const float* __restrict__ wsP,
                                                 float* __restrict__ wsPre,
                                                 float* __restrict__ outC) {
    const int id = (int)blockIdx.x * 256 + (int)threadIdx.x; // b*M + m
    if (id >= B * M) return;
    const float* p = wsP + id * NC * 4;
    float* q = wsPre + id * NC * 4;
    Q pre = {1.f, 0.f, 0.f, 0.f};
#pragma unroll 1
    for (int k = 0; k < NC; ++k) {
        float4 o; o.x = pre.w; o.y = pre.x; o.z = pre.y; o.w = pre.z;
        *(float4*)(q + k * 4) = o;
        float4 pk = *(const float4*)(p + k * 4);
        Q Pk; Pk.w = pk.x; Pk.x = pk.y; Pk.y = pk.z; Pk.z = pk.w;
        pre = qnormalize(qmul(pre, Pk));
    }
    // pre == product of all chunks == C_final[b,m,:]
    float4 cf; cf.x = pre.w; cf.y = pre.x; cf.z = pre.y; cf.w = pre.z;
    *(float4*)(outC + id * 4) = cf;
}

// --------------------------------- phase 3 ---------------------------------
__global__ __launch_bounds__(256) void s3_phase3(const int* __restrict__ tokens,
                                                 const float* __restrict__ embed,
                                                 const float* __restrict__ wsPre,
                                                 float* __restrict__ outSig) {
    __shared__ float lds_g[STEPS_PB * 64]; // 64 KB
    const int b  = (int)blockIdx.x & (B - 1);
    const int cg = (int)blockIdx.x >> 5;
    const int t0 = cg * STEPS_PB;
    const int m  = (int)threadIdx.x & (M - 1);
    const int c  = (int)threadIdx.x >> 4;

    stage_block(tokens + b * T + t0, embed, lds_g);

    const int chunk = cg * CPB + c;
    float4 pp = *(const float4*)(wsPre + (((b * M + m) * NC) + chunk) * 4);
    Q C; C.w = pp.x; C.x = pp.y; C.y = pp.z; C.z = pp.w;

    const float EPS = (float)(1.0 - 1e-7);
    const float* row = lds_g + (c * CL) * 64 + m * 4;
#pragma unroll 1
    for (int i = 0; i < CL; ++i) {
        float4 gv = *(const float4*)(row + i * 64);   // ds_load_b128
        Q g; g.w = gv.x; g.x = gv.y; g.y = gv.z; g.z = gv.w;
        C = qnormalize(qmul(C, qnormalize(g)));
        float f = acosf(fminf(fabsf(C.w), EPS));
        // reduce over the 16 m-lanes (masks 1..8 stay inside each 16-lane
        // half of a wave32)
        f += __shfl_xor(f, 1);
        f += __shfl_xor(f, 2);
        f += __shfl_xor(f, 4);
        f += __shfl_xor(f, 8);
        if (m == 0) outSig[b * T + t0 + c * CL + i] = f * (1.0f / 16.0f);
    }
}

// ------------------------------ entry point --------------------------------
extern "C" void kernel_launch(void* const* d_in, const int* in_sizes, int n_in,
                              void* d_out, int out_size, void* d_ws, size_t ws_size,
                              hipStream_t stream) {
    (void)in_sizes; (void)n_in; (void)out_size; (void)ws_size;
    const int*   tokens = (const int*)d_in[0];     // [B,T] int32
    const float* embed  = (const float*)d_in[1];   // [V,M,4] f32
    float* outC   = (float*)d_out;                 // [B,M,4]
    float* outSig = (float*)d_out + B * M * 4;     // [B,T]
    float* wsP    = (float*)d_ws;                  // [B*M, NC, 4]   (1 MiB)
    float* wsPre  = wsP + B * M * NC * 4;          // [B*M, NC, 4]   (1 MiB)

    s3_phase1<<<dim3(B * (NC / CPB)), dim3(256), 0, stream>>>(tokens, embed, wsP);
    s3_phase2<<<dim3(2), dim3(256), 0, stream>>>(wsP, wsPre, outC);
    s3_phase3<<<dim3(B * (NC / CPB)), dim3(256), 0, stream>>>(tokens, embed, wsPre, outSig);
}